// Head_87033217286245
// MI455X (gfx1250) — compile-verified
//
#include <hip/hip_runtime.h>

typedef __attribute__((ext_vector_type(2))) float v2f;
typedef __attribute__((ext_vector_type(8))) float v8f;

#define BN_EPS 1e-5f

// problem dims
#define B_   128
#define L_   512
#define DB   1024
#define DP   64
#define NS   3
#define HID  512
#define DIN  (DP*NS)      // 192
#define CATN (HID+17)     // 529

// workspace layout (float offsets)
#define WT_OFF   0
#define WT_SZ    (3*DP*DB)          // 196608
#define ZP_OFF   (WT_OFF + WT_SZ)   // zero page for conv boundary rows
#define ZP_SZ    DB                 // 1024 zero floats
#define CONV_OFF (ZP_OFF + ZP_SZ)
#define CONV_SZ  (B_*L_*DP)         // 4194304
#define LOG_OFF  (CONV_OFF + CONV_SZ)
#define LOG_SZ   (B_*L_)
#define EXTR_OFF (LOG_OFF + LOG_SZ)
#define EXTR_SZ  (B_*DIN)
#define H_OFF    (EXTR_OFF + EXTR_SZ)
#define H_SZ     (B_*HID)

// ---------------------------------------------------------------------------
// Kernel 0: transpose conv weights [64,1024,3] -> Wt[3][64][1024] (K contiguous)
// and zero-fill the boundary-row zero page right after it.
// ---------------------------------------------------------------------------
__global__ void k0_transpose_w(const float* __restrict__ conv_w,
                               float* __restrict__ ws0) {
    int idx = blockIdx.x * 256 + threadIdx.x;        // 0 .. WT_SZ+ZP_SZ-1
    if (idx < WT_SZ) {
        int tap = idx / (DP * DB);
        int rem = idx - tap * (DP * DB);
        int n   = rem / DB;
        int c   = rem - n * DB;
        ws0[idx] = conv_w[(n * DB + c) * 3 + tap];
    } else {
        ws0[idx] = 0.0f;                             // zero page
    }
}

// ---------------------------------------------------------------------------
// Kernel 1: conv1d as 3-tap GEMM via V_WMMA_F32_16X16X4_F32.
// Block = 256 threads (8 waves); each wave owns a 16(M) x 64(N) output tile,
// block covers 128 token rows x all 64 channels -> activations streamed once.
// Weights double-buffered in LDS via GLOBAL_LOAD_ASYNC_TO_LDS_B128 (ASYNCcnt),
// row stride padded to 68 floats so lanes hit distinct banks (4n+k mod 64).
// Boundary tap rows read a zero page instead of predicated loads (no EXEC
// manipulation in the inner loop; EXEC stays all-ones around WMMA).
// ---------------------------------------------------------------------------
#define KC    64
#define SBS   68                                    // padded LDS row stride
#define BUF_F (3*DP*SBS)                            // 13056 floats / buffer

__global__ __launch_bounds__(256) void k1_conv_gemm(
    const float* __restrict__ x,      // [B,L,DB]
    const float* __restrict__ wt,     // [3,DP,DB]
    const float* __restrict__ zpage,  // [DB] zeros
    const float* __restrict__ cbias,  // [DP]
    float* __restrict__ conv)         // [B,L,DP]
{
    __shared__ float sB[2 * BUF_F];                 // 104448 bytes

    const int tid    = threadIdx.x;
    const int wave   = tid >> 5;
    const int lane   = tid & 31;
    const int lane16 = lane & 15;
    const int hh     = lane >> 4;

    const int t0 = blockIdx.x * 128;                // token base (inside one batch)
    const int b  = t0 >> 9;
    const int l0 = t0 & (L_ - 1);
    const int lrow = l0 + wave * 16 + lane16;       // A-tile row (tap-center)

    const float* aptr[3];
#pragma unroll
    for (int tap = 0; tap < 3; ++tap) {
        int lt = lrow + tap - 1;
        aptr[tap] = (lt >= 0 && lt < L_) ? (x + ((size_t)b * L_ + lt) * DB) : zpage;
    }

    // async staging: chunk = 3*64 rows x 64 floats = 3072 float4; 12 b128/thread.
    // i = tid + j*256  ->  f4 = tid&15 (const), row = (tid>>4) + 16*j.
    const unsigned sbase = (unsigned)(size_t)(&sB[0]);   // LDS byte offset (addr[31:0])
    const int f4 = tid & 15;
    const int r0 = tid >> 4;
    const float* stage_src0  = wt + (size_t)r0 * DB + f4 * 4;
    const unsigned stage_dst0 = sbase + (unsigned)((r0 * SBS + f4 * 4) * 4);

    auto stage_async = [&](int kc, int buf) {
        const float* s = stage_src0 + kc;
        unsigned d = stage_dst0 + (unsigned)(buf * BUF_F * 4);
#pragma unroll
        for (int j = 0; j < 12; ++j) {
            asm volatile("global_load_async_to_lds_b128 %0, %1, off"
                         :: "v"(d), "v"(s) : "memory");
            s += 16 * DB;
            d += 16 * SBS * 4;
        }
    };

    v8f acc[4];
#pragma unroll
    for (int i = 0; i < 4; ++i)
#pragma unroll
        for (int j = 0; j < 8; ++j) acc[i][j] = 0.0f;

    stage_async(0, 0);
    asm volatile("s_wait_asynccnt 0x0" ::: "memory");
    __syncthreads();

    for (int ci = 0; ci < DB / KC; ++ci) {
        const int kc  = ci * KC;
        const int buf = ci & 1;
        if (ci + 1 < DB / KC) {
            stage_async(kc + KC, buf ^ 1);          // overlap next-chunk weights
#pragma unroll
            for (int tap = 0; tap < 3; ++tap)       // prefetch next-chunk activations
                __builtin_prefetch(aptr[tap] + kc + KC, 0, 0);
        }
        const float* bbuf = &sB[buf * BUF_F];

        for (int k0 = 0; k0 < KC; k0 += 4) {
            const int ka = kc + k0 + 2 * hh;
            v2f a[3];
#pragma unroll
            for (int tap = 0; tap < 3; ++tap)
                a[tap] = *(const v2f*)(aptr[tap] + ka);
#pragma unroll
            for (int tap = 0; tap < 3; ++tap) {
                const float* bb = bbuf + (tap * DP + lane16) * SBS + k0 + 2 * hh;
#pragma unroll
                for (int nt = 0; nt < 4; ++nt) {
                    v2f bm = *(const v2f*)(bb + nt * 16 * SBS);
                    acc[nt] = __builtin_amdgcn_wmma_f32_16x16x4_f32(
                        false, a[tap], false, bm, (short)0, acc[nt], false, false);
                }
            }
        }
        asm volatile("s_wait_asynccnt 0x0" ::: "memory");
        __syncthreads();
    }

    // epilogue: bias + store.  D layout: acc[nt][r] = D[r + 8*hh][nt*16+lane16]
#pragma unroll
    for (int nt = 0; nt < 4; ++nt) {
        const int n = nt * 16 + lane16;
        const float bias = cbias[n];
#pragma unroll
        for (int r = 0; r < 8; ++r) {
            int l = l0 + wave * 16 + r + 8 * hh;
            conv[((size_t)b * L_ + l) * DP + n] = acc[nt][r] + bias;
        }
    }
}

// ---------------------------------------------------------------------------
// Kernel 2: per-token attention logits = dot(conv[t,:], att_w) + att_b
// one wave per token (strided), float2 per lane + xor-shuffle reduce (wave32)
// ---------------------------------------------------------------------------
__global__ void k2_logits(const float* __restrict__ conv,
                          const float* __restrict__ att_w,
                          const float* __restrict__ att_b,
                          float* __restrict__ logits) {
    int gwave = (blockIdx.x * blockDim.x + threadIdx.x) >> 5;   // 0..511
    int lane  = threadIdx.x & 31;
    v2f w = *(const v2f*)(att_w + 2 * lane);
    float ab = att_b[0];
    for (int t = gwave; t < B_ * L_; t += 512) {
        v2f c = *(const v2f*)(conv + (size_t)t * DP + 2 * lane);
        float s = c.x * w.x + c.y * w.y;
        for (int off = 16; off; off >>= 1) s += __shfl_xor(s, off, 32);
        if (lane == 0) logits[t] = s + ab;
    }
}

// ---------------------------------------------------------------------------
// Kernel 3: per (batch, span) masked softmax + weighted sum; BN1 folded in.
// One wave per span; lane owns 2 of the 64 dims.
// ---------------------------------------------------------------------------
__global__ void k3_span(const float* __restrict__ conv,
                        const float* __restrict__ logits,
                        const int*   __restrict__ offsets,
                        const float* __restrict__ bn1_g, const float* __restrict__ bn1_b,
                        const float* __restrict__ bn1_m, const float* __restrict__ bn1_v,
                        float* __restrict__ ext) {
    int bs = blockIdx.x;
    int b = bs / NS, s = bs - b * NS;
    int lane = threadIdx.x;
    int start = offsets[(b * NS + s) * 2 + 0];
    int end   = offsets[(b * NS + s) * 2 + 1];      // inclusive
    const float* lg = logits + b * L_;

    float mloc = -1e30f;
    for (int l = start + lane; l <= end; l += 32) mloc = fmaxf(mloc, lg[l]);
    for (int off = 16; off; off >>= 1) mloc = fmaxf(mloc, __shfl_xor(mloc, off, 32));

    float zloc = 0.0f;
    for (int l = start + lane; l <= end; l += 32) zloc += __expf(lg[l] - mloc);
    for (int off = 16; off; off >>= 1) zloc += __shfl_xor(zloc, off, 32);
    float invz = 1.0f / zloc;

    v2f acc = {0.0f, 0.0f};
    const float* cbase = conv + (size_t)b * L_ * DP + 2 * lane;
    for (int l = start; l <= end; ++l) {
        float w = __expf(lg[l] - mloc) * invz;
        v2f c = *(const v2f*)(cbase + (size_t)l * DP);
        acc.x += w * c.x;
        acc.y += w * c.y;
    }

    int i0 = s * DP + 2 * lane;
    float g0 = bn1_g[i0] * rsqrtf(bn1_v[i0] + BN_EPS);
    float g1 = bn1_g[i0 + 1] * rsqrtf(bn1_v[i0 + 1] + BN_EPS);
    ext[b * DIN + i0]     = (acc.x - bn1_m[i0]) * g0 + bn1_b[i0];
    ext[b * DIN + i0 + 1] = (acc.y - bn1_m[i0 + 1]) * g1 + bn1_b[i0 + 1];
}

// ---------------------------------------------------------------------------
// Kernel 4: FC 192->512 as WMMA f32 GEMM (M=128,N=512,K=192), epilogue:
// +bias, ReLU, then BN2 (h = relu(y)*sc + (beta - mean*sc)).
// 32 blocks x 8 waves; each wave a 16x16 tile.
// ---------------------------------------------------------------------------
__global__ __launch_bounds__(256) void k4_fc(
    const float* __restrict__ ext,   // [128,192] (BN1 already applied)
    const float* __restrict__ fc_w,  // [512,192]
    const float* __restrict__ fc_b,
    const float* __restrict__ bn2_g, const float* __restrict__ bn2_b,
    const float* __restrict__ bn2_m, const float* __restrict__ bn2_v,
    float* __restrict__ h)           // [128,512]
{
    int tid = threadIdx.x;
    int wave = tid >> 5, lane = tid & 31;
    int lane16 = lane & 15, hh = lane >> 4;
    int wid = blockIdx.x * 8 + wave;                // 0..255
    int mt = wid & 7, nt = wid >> 3;                // 8 M-tiles x 32 N-tiles
    int m0 = mt * 16, n0 = nt * 16;

    v8f acc;
#pragma unroll
    for (int j = 0; j < 8; ++j) acc[j] = 0.0f;

    const float* arow = ext  + (size_t)(m0 + lane16) * DIN;
    const float* brow = fc_w + (size_t)(n0 + lane16) * DIN;
    for (int k0 = 0; k0 < DIN; k0 += 4) {
        v2f a  = *(const v2f*)(arow + k0 + 2 * hh);
        v2f bm = *(const v2f*)(brow + k0 + 2 * hh);
        acc = __builtin_amdgcn_wmma_f32_16x16x4_f32(
            false, a, false, bm, (short)0, acc, false, false);
    }

    int j = n0 + lane16;
    float sc   = bn2_g[j] * rsqrtf(bn2_v[j] + BN_EPS);
    float offj = bn2_b[j] - bn2_m[j] * sc;
    float bias = fc_b[j];
#pragma unroll
    for (int r = 0; r < 8; ++r) {
        int row = m0 + r + 8 * hh;
        float y = acc[r] + bias;
        y = fmaxf(y, 0.0f);
        h[(size_t)row * HID + j] = y * sc + offj;
    }
}

// ---------------------------------------------------------------------------
// Kernel 5: final head: out[b,j] = cat(h, in_urls, other) . last_w[j] + last_b[j]
// one wave per batch row.
// ---------------------------------------------------------------------------
__global__ void k5_last(const float* __restrict__ h,
                        const float* __restrict__ in_urls,
                        const float* __restrict__ other,
                        const float* __restrict__ last_w,
                        const float* __restrict__ last_b,
                        float* __restrict__ out) {
    int b = blockIdx.x;
    int lane = threadIdx.x;
    float a0 = 0.f, a1 = 0.f, a2 = 0.f;
    const float* hb = h + (size_t)b * HID;
    for (int i = lane; i < HID; i += 32) {
        float x = hb[i];
        a0 += x * last_w[0 * CATN + i];
        a1 += x * last_w[1 * CATN + i];
        a2 += x * last_w[2 * CATN + i];
    }
    for (int i = HID + lane; i < CATN; i += 32) {
        float x = (i < HID + 3) ? in_urls[b * 3 + (i - HID)]
                                : other[b * 14 + (i - HID - 3)];
        a0 += x * last_w[0 * CATN + i];
        a1 += x * last_w[1 * CATN + i];
        a2 += x * last_w[2 * CATN + i];
    }
    for (int off = 16; off; off >>= 1) {
        a0 += __shfl_xor(a0, off, 32);
        a1 += __shfl_xor(a1, off, 32);
        a2 += __shfl_xor(a2, off, 32);
    }
    if (lane == 0) {
        out[b * 3 + 0] = a0 + last_b[0];
        out[b * 3 + 1] = a1 + last_b[1];
        out[b * 3 + 2] = a2 + last_b[2];
    }
}

// ---------------------------------------------------------------------------
extern "C" void kernel_launch(void* const* d_in, const int* in_sizes, int n_in,
                              void* d_out, int out_size, void* d_ws, size_t ws_size,
                              hipStream_t stream) {
    const float* bert    = (const float*)d_in[0];
    const int*   offsets = (const int*)  d_in[1];
    const float* in_urls = (const float*)d_in[2];
    const float* other   = (const float*)d_in[3];
    const float* conv_w  = (const float*)d_in[4];
    const float* conv_b  = (const float*)d_in[5];
    const float* att_w   = (const float*)d_in[6];
    const float* att_b   = (const float*)d_in[7];
    const float* bn1_g   = (const float*)d_in[8];
    const float* bn1_b   = (const float*)d_in[9];
    const float* bn1_m   = (const float*)d_in[10];
    const float* bn1_v   = (const float*)d_in[11];
    const float* fc_w    = (const float*)d_in[12];
    const float* fc_b    = (const float*)d_in[13];
    const float* bn2_g   = (const float*)d_in[14];
    const float* bn2_b   = (const float*)d_in[15];
    const float* bn2_m   = (const float*)d_in[16];
    const float* bn2_v   = (const float*)d_in[17];
    const float* last_w  = (const float*)d_in[18];
    const float* last_b  = (const float*)d_in[19];

    float* ws     = (float*)d_ws;
    float* wt     = ws + WT_OFF;
    float* zpage  = ws + ZP_OFF;
    float* conv   = ws + CONV_OFF;
    float* logits = ws + LOG_OFF;
    float* ext    = ws + EXTR_OFF;
    float* h      = ws + H_OFF;
    float* out    = (float*)d_out;

    k0_transpose_w<<<(WT_SZ + ZP_SZ) / 256, 256, 0, stream>>>(conv_w, wt);
    k1_conv_gemm<<<(B_ * L_) / 128, 256, 0, stream>>>(bert, wt, zpage, conv_b, conv);
    k2_logits<<<64, 256, 0, stream>>>(conv, att_w, att_b, logits);
    k3_span<<<B_ * NS, 32, 0, stream>>>(conv, logits, offsets,
                                        bn1_g, bn1_b, bn1_m, bn1_v, ext);
    k4_fc<<<32, 256, 0, stream>>>(ext, fc_w, fc_b,
                                  bn2_g, bn2_b, bn2_m, bn2_v, h);
    k5_last<<<B_, 32, 0, stream>>>(h, in_urls, other, last_w, last_b, out);
}